// LRInteractionPredictor_26525718020341
// MI455X (gfx1250) — compile-verified
//
#include <hip/hip_runtime.h>

#define E_EDGES 65536
#define DIM     256
#define NP      8

typedef __attribute__((ext_vector_type(2))) float v2f;
typedef __attribute__((ext_vector_type(8))) float v8f;
typedef __attribute__((ext_vector_type(4))) unsigned int v4u;
typedef __attribute__((ext_vector_type(8))) int v8i;
typedef __attribute__((ext_vector_type(4))) int v4i;

#if defined(__has_builtin)
#if __has_builtin(__builtin_amdgcn_tensor_load_to_lds) && \
    __has_builtin(__builtin_amdgcn_s_wait_tensorcnt)
#define HAVE_TDM 1
#endif
#endif

// -------- workspace layout (floats unless noted) --------
// [0 .. NP*DIM*DIM)      A_p matrices (8 x 256 x 256)
// [WS_C_OFF .. +NP*DIM)  c_p vectors  (8 x 256)
// [WS_INT_OFF ...] as int:
//   ip[0..7]  counts, ip[8..16] padded offsets (poff, poff[8]=total),
//   ip[17..24] cursors, ip[32..] edge ids (-1 = padding slot)
#define WS_C_OFF   (NP * DIM * DIM)
#define WS_INT_OFF (WS_C_OFF + NP * DIM)
#define EID_BASE   32
#define SEG_PAD    64
#define MAX_SLOTS  (E_EDGES + NP * SEG_PAD)
#define LDS_STRIDE 260   // 256 + 4 pad floats -> conflict-free b64 reads

// ---------------- binning ----------------
__global__ void init_kernel(int* ip) {
  int t = blockIdx.x * blockDim.x + threadIdx.x;
  if (t < NP) ip[t] = 0;
  if (t < MAX_SLOTS) ip[EID_BASE + t] = -1;
}

__global__ void count_kernel(const int* __restrict__ idx, int* ip) {
  int e = blockIdx.x * blockDim.x + threadIdx.x;
  if (e < E_EDGES) atomicAdd(&ip[idx[e] & (NP - 1)], 1);
}

__global__ void offsets_kernel(int* ip) {
  if (threadIdx.x == 0 && blockIdx.x == 0) {
    int acc = 0;
    for (int q = 0; q < NP; q++) {
      ip[8 + q]  = acc;
      ip[17 + q] = acc;
      acc += ((ip[q] + SEG_PAD - 1) / SEG_PAD) * SEG_PAD;  // pad to 64
    }
    ip[16] = acc;
  }
}

__global__ void scatter_kernel(const int* __restrict__ idx, int* ip) {
  int e = blockIdx.x * blockDim.x + threadIdx.x;
  if (e < E_EDGES) {
    int p = idx[e] & (NP - 1);
    int pos = atomicAdd(&ip[17 + p], 1);
    ip[EID_BASE + pos] = e;
  }
}

// ---------------- A_p = W_lr[p]^T @ W_bil  (WMMA f32 16x16x4) ----------------
__global__ __launch_bounds__(32) void prep_A_kernel(const float* __restrict__ W_lr,
                                                    const float* __restrict__ W_bil,
                                                    float* __restrict__ A_all) {
  int tile = blockIdx.x;
  int mt = tile & 15, it = (tile >> 4) & 15, p = tile >> 8;
  int lane = threadIdx.x, h = lane >> 4, ml = lane & 15;
  int i0 = it * 16, m0 = mt * 16;

  v8f c = {};
  const float* Wp = W_lr + (size_t)p * DIM * DIM;
  for (int k0 = 0; k0 < DIM; k0 += 4) {
    int j = k0 + 2 * h;
    v2f a, b;
    a.x = Wp[(size_t)j * DIM + i0 + ml];
    a.y = Wp[(size_t)(j + 1) * DIM + i0 + ml];
    b.x = W_bil[(size_t)j * DIM + m0 + ml];
    b.y = W_bil[(size_t)(j + 1) * DIM + m0 + ml];
    c = __builtin_amdgcn_wmma_f32_16x16x4_f32(false, a, false, b, (short)0, c,
                                              false, false);
  }
  float* Ap = A_all + (size_t)p * DIM * DIM;
#pragma unroll
  for (int g = 0; g < 8; g++)
    Ap[(size_t)(i0 + g + 8 * h) * DIM + m0 + ml] = c[g];
}

// ---------------- c_p = W_bil^T @ b_lr[p] ----------------
__global__ void prep_c_kernel(const float* __restrict__ b_lr,
                              const float* __restrict__ W_bil,
                              float* __restrict__ c_all) {
  int t = blockIdx.x * blockDim.x + threadIdx.x;
  int p = t >> 8, m = t & 255;
  float s = 0.f;
  for (int j = 0; j < DIM; j++) s += b_lr[p * DIM + j] * W_bil[j * DIM + m];
  c_all[t] = s;
}

// ---------------- main: 64 same-expert edges / block, 4 waves N-split -------
__global__ __launch_bounds__(128) void main_kernel(const float* __restrict__ z_src,
                                                   const float* __restrict__ z_dst,
                                                   const float* __restrict__ b_bil,
                                                   const float* __restrict__ ws_f,
                                                   const int* __restrict__ ip,
                                                   float* __restrict__ out) {
  __shared__ int   ids[64];
  __shared__ float zs[64 * LDS_STRIDE];   // 64 z_src rows, padded stride
  __shared__ float partial[4][64];

  int base = blockIdx.x * 64;
  if (base >= ip[16]) return;             // uniform over block

  int p = 0;
#pragma unroll
  for (int q = 1; q < NP; q++)
    if (base >= ip[8 + q]) p = q;

  int tid  = threadIdx.x;
  int w    = tid >> 5;                    // wave id 0..3
  int lane = tid & 31;
  int h = lane >> 4, ml = lane & 15;

  if (tid < 64) ids[tid] = ip[EID_BASE + base + tid];

#ifdef HAVE_TDM
  // ---- TDM gather: DMA 16 edge rows (16-bit indices) global -> LDS,
  //      pad 4 DWORDs every 256 DWORDs -> LDS_STRIDE layout. One per wave.
  int idp[8];
#pragma unroll
  for (int r = 0; r < 8; r++) {
    int a0 = ip[EID_BASE + base + w * 16 + 2 * r];
    int a1 = ip[EID_BASE + base + w * 16 + 2 * r + 1];
    idp[r] = __builtin_amdgcn_readfirstlane((a0 & 0xFFFF) | ((a1 & 0xFFFF) << 16));
  }
  unsigned int lds_base =
      (unsigned int)(unsigned long long)(&zs[0]) + (unsigned)(w * 16 * LDS_STRIDE * 4);
  lds_base = (unsigned int)__builtin_amdgcn_readfirstlane((int)lds_base);
  unsigned long long ga = (unsigned long long)z_src;

  v4u g0;
  g0.x = 1u | (1u << 31);                       // count=1, gather_mode=1, 16-bit idx
  g0.y = lds_base;                              // lds_addr
  g0.z = (unsigned int)(ga & 0xFFFFFFFFu);      // global_addr[31:0]
  g0.w = (unsigned int)((ga >> 32) & 0x1FFFFFFu) | (2u << 30);  // addr[56:32], type=2

  v8i g1;
  g1[0] = (2 << 16) | (1 << 20) | (7 << 22) | (3 << 25); // dsz=4B, pad 4dw/256dw
  g1[1] = (256 << 16);   // tensor_dim0 = 256 (low 16 bits at [63:48])
  g1[2] = 0;             // tensor_dim0 hi, tensor_dim1 low16 (65536 -> 0)
  g1[3] = 1 | (256 << 16); // tensor_dim1 hi16 = 1 (65536), tile_dim0 = 256
  g1[4] = 16;            // tile_dim1 = #valid indices
  g1[5] = 256;           // tensor_dim0_stride
  g1[6] = 0;
  g1[7] = 0;

  v4i g2 = {idp[0], idp[1], idp[2], idp[3]};
  v4i g3 = {idp[4], idp[5], idp[6], idp[7]};
  v8i g4 = {0, 0, 0, 0, 0, 0, 0, 0};            // extra operand of 6-arg builtin

  __builtin_amdgcn_tensor_load_to_lds(g0, g1, g2, g3, g4, 0);
  __builtin_amdgcn_s_wait_tensorcnt(0);
  __syncthreads();
#else
  __syncthreads();
#pragma unroll
  for (int i = 0; i < 32; i++) {                 // 4096 float4s / 128 threads
    int idx = tid + 128 * i;
    int r = idx >> 6, c4 = idx & 63;
    int gid = ids[r]; if (gid < 0) gid = 0;
    float4 v = *((const float4*)(z_src + (size_t)gid * DIM) + c4);
    *(float4*)&zs[r * LDS_STRIDE + c4 * 4] = v;
  }
  __syncthreads();
#endif

  const float* Ap   = ws_f + (size_t)p * DIM * DIM;
  const float* cvec = ws_f + WS_C_OFF + p * DIM;

  float acc[4][8];
#pragma unroll
  for (int s = 0; s < 4; s++)
#pragma unroll
    for (int g = 0; g < 8; g++) acc[s][g] = 0.f;

  // wave w owns N-tiles 4w .. 4w+3; one B fragment feeds 4 WMMAs (M-subtiles)
  for (int t = 0; t < 4; t++) {
    int n = (w * 4 + t) * 16 + ml;
    v8f cf0 = {}, cf1 = {}, cf2 = {}, cf3 = {};
    for (int k0 = 0; k0 < DIM; k0 += 4) {
      int j = k0 + 2 * h;
      v2f b;
      b.x = Ap[(size_t)j * DIM + n];
      b.y = Ap[(size_t)(j + 1) * DIM + n];
      v2f a0 = *(const v2f*)&zs[(0 * 16 + ml) * LDS_STRIDE + j];
      v2f a1 = *(const v2f*)&zs[(1 * 16 + ml) * LDS_STRIDE + j];
      v2f a2 = *(const v2f*)&zs[(2 * 16 + ml) * LDS_STRIDE + j];
      v2f a3 = *(const v2f*)&zs[(3 * 16 + ml) * LDS_STRIDE + j];
      cf0 = __builtin_amdgcn_wmma_f32_16x16x4_f32(false, a0, false, b, (short)0, cf0, false, false);
      cf1 = __builtin_amdgcn_wmma_f32_16x16x4_f32(false, a1, false, b, (short)0, cf1, false, false);
      cf2 = __builtin_amdgcn_wmma_f32_16x16x4_f32(false, a2, false, b, (short)0, cf2, false, false);
      cf3 = __builtin_amdgcn_wmma_f32_16x16x4_f32(false, a3, false, b, (short)0, cf3, false, false);
    }
    float cn = cvec[(w * 4 + t) * 16 + ml];
#pragma unroll
    for (int s = 0; s < 4; s++) {
      const v8f& cf = s == 0 ? cf0 : s == 1 ? cf1 : s == 2 ? cf2 : cf3;
#pragma unroll
      for (int g = 0; g < 8; g++) {            // D[M=g+8h, N=ml]
        int M = s * 16 + g + 8 * h;
        int gid = ids[M]; int cg = gid < 0 ? 0 : gid;
        float zd = z_dst[(size_t)cg * DIM + n];
        acc[s][g] += (cf[g] + cn) * zd;
      }
    }
  }

  // reduce across the 16 lanes of each half
#pragma unroll
  for (int s = 0; s < 4; s++)
#pragma unroll
    for (int g = 0; g < 8; g++) {
      float v = acc[s][g];
      v += __shfl_xor(v, 1, 32);
      v += __shfl_xor(v, 2, 32);
      v += __shfl_xor(v, 4, 32);
      v += __shfl_xor(v, 8, 32);
      acc[s][g] = v;
    }

  if (ml == 0) {
#pragma unroll
    for (int s = 0; s < 4; s++)
#pragma unroll
      for (int g = 0; g < 8; g++)
        partial[w][s * 16 + g + 8 * h] = acc[s][g];
  }
  __syncthreads();

  if (tid < 64) {
    float s = partial[0][tid] + partial[1][tid] + partial[2][tid] + partial[3][tid];
    int gid = ids[tid];
    if (gid >= 0) out[gid] = s + b_bil[0];
  }
}

// ---------------- launch ----------------
extern "C" void kernel_launch(void* const* d_in, const int* in_sizes, int n_in,
                              void* d_out, int out_size, void* d_ws, size_t ws_size,
                              hipStream_t stream) {
  const float* z_src = (const float*)d_in[0];
  const float* z_dst = (const float*)d_in[1];
  const int*   idx   = (const int*)d_in[2];
  const float* W_lr  = (const float*)d_in[3];
  const float* b_lr  = (const float*)d_in[4];
  const float* W_bil = (const float*)d_in[5];
  const float* b_bil = (const float*)d_in[6];
  float* out  = (float*)d_out;
  float* ws_f = (float*)d_ws;
  int*   ip   = (int*)(ws_f + WS_INT_OFF);

  init_kernel<<<(MAX_SLOTS + 255) / 256, 256, 0, stream>>>(ip);
  count_kernel<<<E_EDGES / 256, 256, 0, stream>>>(idx, ip);
  offsets_kernel<<<1, 32, 0, stream>>>(ip);
  scatter_kernel<<<E_EDGES / 256, 256, 0, stream>>>(idx, ip);
  prep_A_kernel<<<(DIM / 16) * (DIM / 16) * NP, 32, 0, stream>>>(W_lr, W_bil, ws_f);
  prep_c_kernel<<<(NP * DIM) / 256, 256, 0, stream>>>(b_lr, W_bil, ws_f + WS_C_OFF);
  main_kernel<<<E_EDGES / 64 + NP, 128, 0, stream>>>(z_src, z_dst, b_bil, ws_f, ip, out);
}